// MyGraphAttn_11622181503638
// MI455X (gfx1250) — compile-verified
//
#include <hip/hip_runtime.h>

typedef __attribute__((ext_vector_type(2))) float v2f;
typedef __attribute__((ext_vector_type(8))) float v8f;

#define IN_DIM 256
#define HID 64
#define HEADS 2
#define HIDH (HEADS * HID)   // 128
#define OUT_DIM 256
#define NEG_SLOPE 0.2f

#if __has_builtin(__builtin_amdgcn_global_load_async_to_lds_b128) && \
    __has_builtin(__builtin_amdgcn_s_wait_asynccnt)
#define GAT_HAVE_ASYNC_LDS 1
// parameter type per hipcc diagnostic: AS1/AS3 pointers to gcc-vector int4
typedef int gat_v4i __attribute__((vector_size(4 * sizeof(int))));
typedef __attribute__((address_space(1))) gat_v4i gat_gv4i;
typedef __attribute__((address_space(3))) gat_v4i gat_lv4i;
#endif

// ---- order-preserving float<->int map for atomic max on floats ----
__device__ __forceinline__ int f2ord(float f) {
    int i = __float_as_int(f);
    return (i >= 0) ? i : (i ^ 0x7fffffff);
}
__device__ __forceinline__ float ord2f(int i) {
    return __int_as_float((i >= 0) ? i : (i ^ 0x7fffffff));
}

// ---------------- init: zero agg & s, set m_ord = ord(-inf) ----------------
__global__ void gat_init_kernel(float* __restrict__ agg, float* __restrict__ s,
                                int* __restrict__ mo, int N) {
    int tid = blockIdx.x * blockDim.x + threadIdx.x;
    int stride = gridDim.x * blockDim.x;
    long long total = (long long)N * HIDH;
    for (long long i = tid; i < total; i += stride) agg[i] = 0.0f;
    int t2 = N * HEADS;
    int neg_inf_ord = f2ord(-__builtin_inff());
    for (int i = tid; i < t2; i += stride) { s[i] = 0.0f; mo[i] = neg_inf_ord; }
}

// ---------------- GEMM1: h[N,128] = feat[N,256] @ W[256,128] (f32 WMMA) ----
// Block = 256 threads = 8 waves; block owns 16 rows, wave w owns cols [16w,16w+16)
__global__ __launch_bounds__(256) void gat_gemm1_kernel(
        const float* __restrict__ feat, const float* __restrict__ W,
        float* __restrict__ h, int N) {
    __shared__ float Alds[16 * 260];   // padded stride: bank-conflict-free
    const int row0 = blockIdx.x * 16;
    const int t = threadIdx.x;

    // stage 16x256 A tile (16 threads/row, 4 x 16B each)
    {
        int r = t >> 4;
        int cbase = (t & 15) * 16;
        int grow = row0 + r; if (grow >= N) grow = N - 1;
        const float* gsrc = feat + (size_t)grow * IN_DIM + cbase;
        float* ldst = &Alds[r * 260 + cbase];
#ifdef GAT_HAVE_ASYNC_LDS
        // async DMA global->LDS: same 16B stride on both sides, shared offset
        __builtin_amdgcn_global_load_async_to_lds_b128(
            (gat_gv4i*)gsrc, (gat_lv4i*)ldst, 0, 0);
        __builtin_amdgcn_global_load_async_to_lds_b128(
            (gat_gv4i*)gsrc, (gat_lv4i*)ldst, 16, 0);
        __builtin_amdgcn_global_load_async_to_lds_b128(
            (gat_gv4i*)gsrc, (gat_lv4i*)ldst, 32, 0);
        __builtin_amdgcn_global_load_async_to_lds_b128(
            (gat_gv4i*)gsrc, (gat_lv4i*)ldst, 48, 0);
        __builtin_amdgcn_s_wait_asynccnt(0);
#else
        #pragma unroll
        for (int j = 0; j < 4; ++j)
            *(float4*)(ldst + j * 4) = *(const float4*)(gsrc + j * 4);
#endif
    }
    __syncthreads();

    const int lane = t & 31;
    const int wave = t >> 5;
    const int col0 = wave * 16;
    const int n = col0 + (lane & 15);
    const int hi = lane >> 4;                 // K sub-offset selector
    const float* Arow = &Alds[(lane & 15) * 260];

    v8f c = {};
    #pragma unroll 8
    for (int kk = 0; kk < IN_DIM; kk += 4) {
        v2f a, b;
        a.x = Arow[kk + 2 * hi];
        a.y = Arow[kk + 2 * hi + 1];
        b.x = W[(size_t)(kk + 2 * hi) * HIDH + n];
        b.y = W[(size_t)(kk + 2 * hi + 1) * HIDH + n];
        c = __builtin_amdgcn_wmma_f32_16x16x4_f32(false, a, false, b,
                                                  (short)0, c, false, false);
    }
    // epilogue: uniform fast path (no per-element exec masking)
    if (row0 + 16 <= N) {
        float* hp = h + (size_t)(row0 + 8 * hi) * HIDH + n;
        #pragma unroll
        for (int v = 0; v < 8; ++v) hp[(size_t)v * HIDH] = c[v];
    } else {
        #pragma unroll
        for (int v = 0; v < 8; ++v) {
            int row = row0 + v + 8 * hi;
            if (row < N) h[(size_t)row * HIDH + n] = c[v];
        }
    }
}

// ---------------- el/er: per (node, head) dot with attn vectors ----------------
__global__ void gat_attn_dot_kernel(const float* __restrict__ h,
                                    const float* __restrict__ attn_l,
                                    const float* __restrict__ attn_r,
                                    float* __restrict__ el, float* __restrict__ er,
                                    int N) {
    int tid = blockIdx.x * blockDim.x + threadIdx.x;
    int stride = gridDim.x * blockDim.x;
    int total = N * HEADS;
    for (int idx = tid; idx < total; idx += stride) {
        int node = idx >> 1, hd = idx & 1;
        const float* hp = h + (size_t)node * HIDH + hd * HID;
        const float* al = attn_l + hd * HID;
        const float* ar = attn_r + hd * HID;
        float sl = 0.0f, sr = 0.0f;
        #pragma unroll 8
        for (int d = 0; d < HID; ++d) { float v = hp[d]; sl += v * al[d]; sr += v * ar[d]; }
        el[idx] = sl; er[idx] = sr;
    }
}

__device__ __forceinline__ float leaky(float x) {
    return x > 0.0f ? x : NEG_SLOPE * x;
}

// ---------------- segment max of leaky(el[src]+er[dst]) over dst ----------------
__global__ void gat_edge_max_kernel(const int* __restrict__ src, const int* __restrict__ dst,
                                    const float* __restrict__ el, const float* __restrict__ er,
                                    int* __restrict__ mo, int E) {
    int tid = blockIdx.x * blockDim.x + threadIdx.x;
    int stride = gridDim.x * blockDim.x;
    for (int e = tid; e < E; e += stride) {
        int s_ = src[e], d_ = dst[e];
        #pragma unroll
        for (int hd = 0; hd < HEADS; ++hd) {
            float x = leaky(el[s_ * HEADS + hd] + er[d_ * HEADS + hd]);
            atomicMax(&mo[d_ * HEADS + hd], f2ord(x));
        }
    }
}

// ---------------- ee = exp(e - m[dst]); s[dst] += ee ----------------
__global__ void gat_edge_exp_kernel(const int* __restrict__ src, const int* __restrict__ dst,
                                    const float* __restrict__ el, const float* __restrict__ er,
                                    const int* __restrict__ mo, float* __restrict__ ee,
                                    float* __restrict__ s, int E) {
    int tid = blockIdx.x * blockDim.x + threadIdx.x;
    int stride = gridDim.x * blockDim.x;
    for (int e = tid; e < E; e += stride) {
        int s_ = src[e], d_ = dst[e];
        #pragma unroll
        for (int hd = 0; hd < HEADS; ++hd) {
            float x = leaky(el[s_ * HEADS + hd] + er[d_ * HEADS + hd]);
            float m = ord2f(mo[d_ * HEADS + hd]);
            float v = __expf(x - m);
            ee[(size_t)e * HEADS + hd] = v;
            atomicAdd(&s[d_ * HEADS + hd], v);
        }
    }
}

// ---------------- agg[dst] += alpha * h[src]; wave per edge, lane = 1 float4 ----
__global__ __launch_bounds__(256) void gat_aggregate_kernel(
        const int* __restrict__ src, const int* __restrict__ dst,
        const float* __restrict__ ee, const float* __restrict__ s,
        const float* __restrict__ h, float* __restrict__ agg, int E) {
    int gtid = blockIdx.x * blockDim.x + threadIdx.x;
    int wid = gtid >> 5;
    int lane = gtid & 31;
    int nwaves = (gridDim.x * blockDim.x) >> 5;
    int hd = lane >> 4;                // dims [lane*4, lane*4+4) all in one head
    for (int e = wid; e < E; e += nwaves) {
        int s_ = src[e], d_ = dst[e];
        // prefetch next edge's gathered row to hide latency behind the atomics
        int enext = e + nwaves;
        if (enext < E) {
            int sn = src[enext];
            __builtin_prefetch(h + (size_t)sn * HIDH + lane * 4, 0, 1);
        }
        float alpha = ee[(size_t)e * HEADS + hd] / s[d_ * HEADS + hd];
        float4 v = *(const float4*)(h + (size_t)s_ * HIDH + lane * 4);
        float* ap = agg + (size_t)d_ * HIDH + lane * 4;
        atomicAdd(ap + 0, alpha * v.x);
        atomicAdd(ap + 1, alpha * v.y);
        atomicAdd(ap + 2, alpha * v.z);
        atomicAdd(ap + 3, alpha * v.w);
    }
}

// ---------------- GEMM2: out = relu(relu(agg + gat_bias) @ lin_W + lin_b) ------
// Block = 512 threads = 16 waves; block owns 16 rows, wave w owns cols [16w,16w+16)
__global__ __launch_bounds__(512) void gat_gemm2_kernel(
        const float* __restrict__ agg, const float* __restrict__ gat_bias,
        const float* __restrict__ lin_W, const float* __restrict__ lin_b,
        float* __restrict__ out, int N) {
    __shared__ float Alds[16 * 132];
    const int row0 = blockIdx.x * 16;
    const int t = threadIdx.x;

    // stage relu(agg + gat_bias) tile: 16 rows x 128, one float4 per thread
    {
        int r = t >> 5;
        int c4 = (t & 31) * 4;
        int grow = row0 + r; if (grow >= N) grow = N - 1;
        float4 v = *(const float4*)(agg + (size_t)grow * HIDH + c4);
        float4 b = *(const float4*)(gat_bias + c4);
        v.x = fmaxf(v.x + b.x, 0.0f);
        v.y = fmaxf(v.y + b.y, 0.0f);
        v.z = fmaxf(v.z + b.z, 0.0f);
        v.w = fmaxf(v.w + b.w, 0.0f);
        *(float4*)(&Alds[r * 132 + c4]) = v;
    }
    __syncthreads();

    const int lane = t & 31;
    const int wave = t >> 5;
    const int col0 = wave * 16;
    const int n = col0 + (lane & 15);
    const int hi = lane >> 4;
    const float* Arow = &Alds[(lane & 15) * 132];

    v8f c = {};
    #pragma unroll 8
    for (int kk = 0; kk < HIDH; kk += 4) {
        v2f a, b;
        a.x = Arow[kk + 2 * hi];
        a.y = Arow[kk + 2 * hi + 1];
        b.x = lin_W[(size_t)(kk + 2 * hi) * OUT_DIM + n];
        b.y = lin_W[(size_t)(kk + 2 * hi + 1) * OUT_DIM + n];
        c = __builtin_amdgcn_wmma_f32_16x16x4_f32(false, a, false, b,
                                                  (short)0, c, false, false);
    }
    float bias = lin_b[n];
    if (row0 + 16 <= N) {
        float* op = out + (size_t)(row0 + 8 * hi) * OUT_DIM + n;
        #pragma unroll
        for (int v = 0; v < 8; ++v) op[(size_t)v * OUT_DIM] = fmaxf(c[v] + bias, 0.0f);
    } else {
        #pragma unroll
        for (int v = 0; v < 8; ++v) {
            int row = row0 + v + 8 * hi;
            if (row < N) out[(size_t)row * OUT_DIM + n] = fmaxf(c[v] + bias, 0.0f);
        }
    }
}

extern "C" void kernel_launch(void* const* d_in, const int* in_sizes, int n_in,
                              void* d_out, int out_size, void* d_ws, size_t ws_size,
                              hipStream_t stream) {
    const float* feat     = (const float*)d_in[0];
    const int*   src      = (const int*)  d_in[1];
    const int*   dst      = (const int*)  d_in[2];
    const float* W        = (const float*)d_in[3];
    const float* attn_l   = (const float*)d_in[4];
    const float* attn_r   = (const float*)d_in[5];
    const float* gat_bias = (const float*)d_in[6];
    const float* lin_W    = (const float*)d_in[7];
    const float* lin_b    = (const float*)d_in[8];
    float* out = (float*)d_out;

    const int N = in_sizes[0] / IN_DIM;
    const int E = in_sizes[1];

    // workspace carve-up (~118 MB)
    float* h    = (float*)d_ws;                       // N * 128
    float* el   = h  + (size_t)N * HIDH;              // N * 2
    float* er   = el + (size_t)N * HEADS;             // N * 2
    int*   mo   = (int*)(er + (size_t)N * HEADS);     // N * 2 (ordered-int max)
    float* sden = (float*)(mo + (size_t)N * HEADS);   // N * 2
    float* ee   = sden + (size_t)N * HEADS;           // E * 2
    float* agg  = ee + (size_t)E * HEADS;             // N * 128

    const int nRowTiles = (N + 15) / 16;

    gat_init_kernel<<<2048, 256, 0, stream>>>(agg, sden, mo, N);
    gat_gemm1_kernel<<<nRowTiles, 256, 0, stream>>>(feat, W, h, N);
    gat_attn_dot_kernel<<<1024, 256, 0, stream>>>(h, attn_l, attn_r, el, er, N);
    gat_edge_max_kernel<<<2048, 256, 0, stream>>>(src, dst, el, er, mo, E);
    gat_edge_exp_kernel<<<2048, 256, 0, stream>>>(src, dst, el, er, mo, ee, sden, E);
    gat_aggregate_kernel<<<4096, 256, 0, stream>>>(src, dst, ee, sden, h, agg, E);
    gat_gemm2_kernel<<<nRowTiles, 512, 0, stream>>>(agg, gat_bias, lin_W, lin_b, out, N);
}